// GATv2_2147483648706
// MI455X (gfx1250) — compile-verified
//
#include <hip/hip_runtime.h>
#include <math.h>

// GIN + attention-pool + critic for MI455X (gfx1250).
// Dense 16x32 / 32x32 linears use V_WMMA_F32_16X16X4_F32 (fp32, full precision).
// Edge aggregation uses global_atomic_add_f32 (agg arrays fit in 192MB L2),
// with 4/8 lanes per edge so gather loads coalesce into contiguous rows, and
// global_prefetch_b8 to warm L2 ahead of the random gather.
// Wave32: one 16-node tile per wave, 8 waves (256 threads) per block.

typedef __attribute__((ext_vector_type(2))) float v2f;
typedef __attribute__((ext_vector_type(8))) float v8f;

#define NN 262144   // N_NODES
#define NE 4194304  // N_EDGES
#define NG 1024     // N_GRAPHS
#define LSTR 34     // LDS tile row stride (floats), keeps rows 8B aligned

__device__ __forceinline__ v8f v8f_zero() {
  v8f z = {0.f, 0.f, 0.f, 0.f, 0.f, 0.f, 0.f, 0.f};
  return z;
}

// D = A(16xK) * B(Kx16) + C, K step of 4, fp32.
__device__ __forceinline__ v8f wmma4(v2f a, v2f b, v8f c) {
  return __builtin_amdgcn_wmma_f32_16x16x4_f32(
      /*neg_a=*/false, a, /*neg_b=*/false, b,
      /*c_mod=*/(short)0, c, /*reuse_a=*/false, /*reuse_b=*/false);
}

// B tile (4x16) from row-major weight W[K][32]; tile selects cols [16t,16t+16).
// Low lanes hold rows 4s..4s+1, high lanes rows 4s+2..4s+3 (col = lane&15).
__device__ __forceinline__ v2f load_b_tile(const float* __restrict__ W, int kstep,
                                           int tile, int lane) {
  int col = (lane & 15) + (tile << 4);
  int krow = (kstep << 2) + ((lane >> 4) << 1);
  v2f b;
  b.x = W[krow * 32 + col];
  b.y = W[(krow + 1) * 32 + col];
  return b;
}

// A layout (f32 16x4 per step): row = lane&15; lanes<16 hold K=4s,4s+1,
// lanes>=16 hold K=4s+2,4s+3.
__device__ __forceinline__ void load_a_sum(const float* __restrict__ X,
                                           const float* __restrict__ G, int base,
                                           int stride, v2f* a, int ksteps, int lane) {
  int r = lane & 15, hi = lane >> 4;
  const float* xr = X + (size_t)(base + r) * stride;
  const float* gr = G + (size_t)(base + r) * stride;
#pragma unroll
  for (int s = 0; s < 8; ++s) {
    if (s >= ksteps) break;
    int c = (s << 2) + (hi << 1);
    a[s].x = xr[c] + gr[c];
    a[s].y = xr[c + 1] + gr[c + 1];
  }
}

__device__ __forceinline__ void load_a_global32(const float* __restrict__ X, int base,
                                                v2f (&a)[8], int lane) {
  int r = lane & 15, hi = lane >> 4;
  const float* xr = X + (size_t)(base + r) * 32;
#pragma unroll
  for (int s = 0; s < 8; ++s) {
    int c = (s << 2) + (hi << 1);
    a[s].x = xr[c];
    a[s].y = xr[c + 1];
  }
}

__device__ __forceinline__ void load_a_lds(const float* __restrict__ tile,
                                           v2f (&a)[8], int lane) {
  int r = lane & 15, hi = lane >> 4;
#pragma unroll
  for (int s = 0; s < 8; ++s) {
    int c = (s << 2) + (hi << 1);
    a[s].x = tile[r * LSTR + c];
    a[s].y = tile[r * LSTR + c + 1];
  }
}

template <int KSTEPS>
__device__ __forceinline__ void linear_tiles(const v2f (&a)[KSTEPS],
                                             const float* __restrict__ W, v8f& c0,
                                             v8f& c1, int lane) {
#pragma unroll
  for (int s = 0; s < KSTEPS; ++s) {
    c0 = wmma4(a[s], load_b_tile(W, s, 0, lane), c0);
    c1 = wmma4(a[s], load_b_tile(W, s, 1, lane), c1);
  }
}

// C layout: VGPR j <-> row j + 8*(lane>=16), col = lane&15 (tile1 cols +16).
__device__ __forceinline__ void store_c_lds_relu(float* __restrict__ tile, v8f c0,
                                                 v8f c1, const float* __restrict__ bias,
                                                 int lane) {
  int col = lane & 15, hi = lane >> 4;
  float b0 = bias[col], b1 = bias[col + 16];
#pragma unroll
  for (int j = 0; j < 8; ++j) {
    int row = j + (hi << 3);
    tile[row * LSTR + col] = fmaxf(c0[j] + b0, 0.f);
    tile[row * LSTR + col + 16] = fmaxf(c1[j] + b1, 0.f);
  }
}

__device__ __forceinline__ void store_c_global_relu(float* __restrict__ H, int base,
                                                    v8f c0, v8f c1,
                                                    const float* __restrict__ bias,
                                                    int lane) {
  int col = lane & 15, hi = lane >> 4;
  float b0 = bias[col], b1 = bias[col + 16];
#pragma unroll
  for (int j = 0; j < 8; ++j) {
    size_t row = (size_t)(base + j + (hi << 3));
    H[row * 32 + col] = fmaxf(c0[j] + b0, 0.f);
    H[row * 32 + col + 16] = fmaxf(c1[j] + b1, 0.f);
  }
}

__device__ __forceinline__ int lower_bound_i(const int* __restrict__ a, int n,
                                             int key) {
  int lo = 0, hi = n;
  while (lo < hi) {
    int m = (lo + hi) >> 1;
    if (a[m] < key) lo = m + 1; else hi = m;
  }
  return lo;
}

// ---------------- kernels ----------------

__global__ void k_zero4(float4* __restrict__ p, int n4) {
  int i = blockIdx.x * blockDim.x + threadIdx.x;
  if (i < n4) p[i] = make_float4(0.f, 0.f, 0.f, 0.f);
}

// LPE lanes cooperate on one edge; each lane moves one float4 chunk of the
// C = 4*LPE channel row. Consecutive lanes touch consecutive addresses, so the
// gather load and the atomics coalesce into whole-row (64B/128B) requests.
template <int LPE>
__global__ void k_scatter(const float* __restrict__ feat, const int* __restrict__ src,
                          const int* __restrict__ dst, float* __restrict__ agg,
                          int n_edges) {
  const int C = 4 * LPE;
  int gid = blockIdx.x * blockDim.x + threadIdx.x;
  int e = gid / LPE;
  int chunk = gid % LPE;
  if (e >= n_edges) return;
  int s = src[e], d = dst[e];
  // warm L2 for a future random gather row (global_prefetch_b8)
  if (chunk == 0) {
    int ep = e + 4096;
    if (ep < n_edges)
      __builtin_prefetch(feat + (size_t)src[ep] * C, 0, 1);
  }
  const float4 v = *(const float4*)(feat + (size_t)s * C + chunk * 4);
  float* ad = agg + (size_t)d * C + chunk * 4;
  atomicAdd(ad + 0, v.x);
  atomicAdd(ad + 1, v.y);
  atomicAdd(ad + 2, v.z);
  atomicAdd(ad + 3, v.w);
}

// h1 = relu((x + agg1) @ W1[16x32] + b1)
__global__ void k_gin_lin16(const float* __restrict__ x, const float* __restrict__ agg,
                            const float* __restrict__ W, const float* __restrict__ b,
                            float* __restrict__ out) {
  int lane = threadIdx.x & 31, wave = threadIdx.x >> 5;
  int base = (blockIdx.x * 8 + wave) * 16;
  v2f a[4];
  load_a_sum(x, agg, base, 16, a, 4, lane);
  v8f c0 = v8f_zero(), c1 = v8f_zero();
  linear_tiles<4>(a, W, c0, c1, lane);
  store_c_global_relu(out, base, c0, c1, b, lane);
}

// h = relu((hin + agg) @ W[32x32] + b)
__global__ void k_gin_lin32(const float* __restrict__ hin, const float* __restrict__ agg,
                            const float* __restrict__ W, const float* __restrict__ b,
                            float* __restrict__ out) {
  int lane = threadIdx.x & 31, wave = threadIdx.x >> 5;
  int base = (blockIdx.x * 8 + wave) * 16;
  v2f a[8];
  load_a_sum(hin, agg, base, 32, a, 8, lane);
  v8f c0 = v8f_zero(), c1 = v8f_zero();
  linear_tiles<8>(a, W, c0, c1, lane);
  store_c_global_relu(out, base, c0, c1, b, lane);
}

// Fused gate MLP (32->32->32->1) and transform MLP (32->32->32), reading h once.
__global__ void k_gate_t(const float* __restrict__ h, const float* __restrict__ gw1,
                         const float* __restrict__ gb1, const float* __restrict__ gw2,
                         const float* __restrict__ gb2, const float* __restrict__ gw3,
                         const float* __restrict__ gb3, const float* __restrict__ aw1,
                         const float* __restrict__ ab1, const float* __restrict__ aw2,
                         const float* __restrict__ ab2, float* __restrict__ gate,
                         float* __restrict__ t_out) {
  __shared__ float tile[8][16 * LSTR];
  int lane = threadIdx.x & 31, wave = threadIdx.x >> 5;
  float* my = tile[wave];
  int base = (blockIdx.x * 8 + wave) * 16;

  v2f ah[8];
  load_a_global32(h, base, ah, lane);  // kept live for both branches

  // gate branch: relu(h@gw1+gb1) -> relu(@gw2+gb2) -> @gw3+gb3
  v8f c0 = v8f_zero(), c1 = v8f_zero();
  linear_tiles<8>(ah, gw1, c0, c1, lane);
  store_c_lds_relu(my, c0, c1, gb1, lane);
  __syncthreads();
  v2f ag[8];
  load_a_lds(my, ag, lane);
  __syncthreads();
  c0 = v8f_zero(); c1 = v8f_zero();
  linear_tiles<8>(ag, gw2, c0, c1, lane);
  store_c_lds_relu(my, c0, c1, gb2, lane);
  __syncthreads();
  {
    int n = lane & 15, hi = lane >> 4;
    float p = 0.f;
#pragma unroll
    for (int k = 0; k < 16; ++k) p += my[n * LSTR + (hi << 4) + k] * gw3[(hi << 4) + k];
    p += __shfl_xor(p, 16, 32);
    if (lane < 16) gate[base + n] = p + gb3[0];
  }
  __syncthreads();

  // transform branch: relu(h@aw1+ab1) -> relu(@aw2+ab2)
  c0 = v8f_zero(); c1 = v8f_zero();
  linear_tiles<8>(ah, aw1, c0, c1, lane);
  store_c_lds_relu(my, c0, c1, ab1, lane);
  __syncthreads();
  v2f at[8];
  load_a_lds(my, at, lane);
  c0 = v8f_zero(); c1 = v8f_zero();
  linear_tiles<8>(at, aw2, c0, c1, lane);
  store_c_global_relu(t_out, base, c0, c1, ab2, lane);
}

// One block per graph: segment softmax of gate over sorted batch_vec, then
// pooled[g] = sum attn * t.
__global__ void k_pool(const float* __restrict__ gate, const float* __restrict__ t,
                       const int* __restrict__ batch, int n_nodes,
                       float* __restrict__ pooled) {
  int g = blockIdx.x;
  int tid = threadIdx.x;
  __shared__ float red[256];
  __shared__ int range[2];
  if (tid == 0) {
    range[0] = lower_bound_i(batch, n_nodes, g);
    range[1] = lower_bound_i(batch, n_nodes, g + 1);
  }
  __syncthreads();
  int s = range[0], e = range[1];
  if (s >= e) {  // empty graph: pooled row = 0
    if (tid < 32) pooled[(size_t)g * 32 + tid] = 0.f;
    return;
  }
  // segment max
  float m = -INFINITY;
  for (int i = s + tid; i < e; i += 256) m = fmaxf(m, gate[i]);
  red[tid] = m;
  __syncthreads();
  for (int w = 128; w > 0; w >>= 1) {
    if (tid < w) red[tid] = fmaxf(red[tid], red[tid + w]);
    __syncthreads();
  }
  float gmax = red[0];
  __syncthreads();
  // segment sum of exp
  float sum = 0.f;
  for (int i = s + tid; i < e; i += 256) sum += expf(gate[i] - gmax);
  red[tid] = sum;
  __syncthreads();
  for (int w = 128; w > 0; w >>= 1) {
    if (tid < w) red[tid] += red[tid + w];
    __syncthreads();
  }
  float denom = fmaxf(red[0], 1e-16f);
  __syncthreads();
  // weighted pooling: 8 node-groups x 32 channels
  int grp = tid >> 5, ch = tid & 31;
  float acc = 0.f;
  for (int i = s + grp; i < e; i += 8) {
    float w = expf(gate[i] - gmax) / denom;
    acc += w * t[(size_t)i * 32 + ch];
  }
  red[tid] = acc;
  __syncthreads();
  if (grp == 0) {
    float tot = 0.f;
#pragma unroll
    for (int k = 0; k < 8; ++k) tot += red[k * 32 + ch];
    pooled[(size_t)g * 32 + ch] = tot;
  }
}

// critic_ff over pooled[G][32]: WMMA tiles of 16 graphs.
__global__ void k_critic(const float* __restrict__ pooled, const float* __restrict__ fw1,
                         const float* __restrict__ fb1, const float* __restrict__ fw2,
                         const float* __restrict__ fb2, const float* __restrict__ fw3,
                         const float* __restrict__ fb3, float* __restrict__ out) {
  __shared__ float tile[8][16 * LSTR];
  int lane = threadIdx.x & 31, wave = threadIdx.x >> 5;
  float* my = tile[wave];
  int base = (blockIdx.x * 8 + wave) * 16;  // graph tile base
  v2f a0[8];
  load_a_global32(pooled, base, a0, lane);
  v8f c0 = v8f_zero(), c1 = v8f_zero();
  linear_tiles<8>(a0, fw1, c0, c1, lane);
  store_c_lds_relu(my, c0, c1, fb1, lane);
  __syncthreads();
  v2f a1[8];
  load_a_lds(my, a1, lane);
  c0 = v8f_zero(); c1 = v8f_zero();
  linear_tiles<8>(a1, fw2, c0, c1, lane);
  store_c_lds_relu(my, c0, c1, fb2, lane);
  __syncthreads();
  int n = lane & 15, hi = lane >> 4;
  float p = 0.f;
#pragma unroll
  for (int k = 0; k < 16; ++k) p += my[n * LSTR + (hi << 4) + k] * fw3[(hi << 4) + k];
  p += __shfl_xor(p, 16, 32);
  if (lane < 16) out[base + n] = p + fb3[0];
}

// ---------------- launcher ----------------

extern "C" void kernel_launch(void* const* d_in, const int* in_sizes, int n_in,
                              void* d_out, int out_size, void* d_ws, size_t ws_size,
                              hipStream_t stream) {
  const float* x = (const float*)d_in[0];
  const float* w1 = (const float*)d_in[1];
  const float* b1 = (const float*)d_in[2];
  const float* w2 = (const float*)d_in[3];
  const float* b2 = (const float*)d_in[4];
  const float* gw1 = (const float*)d_in[5];
  const float* gb1 = (const float*)d_in[6];
  const float* gw2 = (const float*)d_in[7];
  const float* gb2 = (const float*)d_in[8];
  const float* gw3 = (const float*)d_in[9];
  const float* gb3 = (const float*)d_in[10];
  const float* aw1 = (const float*)d_in[11];
  const float* ab1 = (const float*)d_in[12];
  const float* aw2 = (const float*)d_in[13];
  const float* ab2 = (const float*)d_in[14];
  const float* fw1 = (const float*)d_in[15];
  const float* fb1 = (const float*)d_in[16];
  const float* fw2 = (const float*)d_in[17];
  const float* fb2 = (const float*)d_in[18];
  const float* fw3 = (const float*)d_in[19];
  const float* fb3 = (const float*)d_in[20];
  const int* edge_index = (const int*)d_in[21];
  const int* src = edge_index;
  const int* dst = edge_index + NE;
  const int* batch = (const int*)d_in[22];
  float* out = (float*)d_out;

  // workspace layout (floats), with reuse after liveness ends
  float* ws = (float*)d_ws;
  float* agg1 = ws;                          // N*16, live through lin16
  float* h1 = ws + (size_t)NN * 16;          // N*32, live through lin32
  float* agg2 = ws + (size_t)NN * 48;        // N*32, live through lin32
  float* hbuf = ws + (size_t)NN * 80;        // N*32, live through gate/t
  float* gate = agg1;                        // N     (reuse agg1)
  float* tbuf = h1;                          // N*32  (reuse h1)
  float* pooled = agg2;                      // G*32  (reuse agg2)

  k_zero4<<<(NN * 16 / 4 + 255) / 256, 256, 0, stream>>>((float4*)agg1, NN * 16 / 4);
  k_zero4<<<(NN * 32 / 4 + 255) / 256, 256, 0, stream>>>((float4*)agg2, NN * 32 / 4);

  // layer 1: 4 lanes/edge (16 ch), layer 2: 8 lanes/edge (32 ch)
  k_scatter<4><<<(size_t)NE * 4 / 256, 256, 0, stream>>>(x, src, dst, agg1, NE);
  k_gin_lin16<<<NN / 128, 256, 0, stream>>>(x, agg1, w1, b1, h1);

  k_scatter<8><<<(size_t)NE * 8 / 256, 256, 0, stream>>>(h1, src, dst, agg2, NE);
  k_gin_lin32<<<NN / 128, 256, 0, stream>>>(h1, agg2, w2, b2, hbuf);

  k_gate_t<<<NN / 128, 256, 0, stream>>>(hbuf, gw1, gb1, gw2, gb2, gw3, gb3, aw1,
                                         ab1, aw2, ab2, gate, tbuf);

  k_pool<<<NG, 256, 0, stream>>>(gate, tbuf, batch, NN, pooled);

  k_critic<<<NG / 128, 256, 0, stream>>>(pooled, fw1, fb1, fw2, fb2, fw3, fb3, out);

  (void)in_sizes; (void)n_in; (void)out_size; (void)ws_size;
}